// SoftTarget_Ours_53145925321221
// MI455X (gfx1250) — compile-verified
//
#include <hip/hip_runtime.h>
#include <cstdint>

#define ASPC(n) __attribute__((address_space(n)))
typedef int v4i __attribute__((vector_size(4 * sizeof(int))));

#if defined(__has_builtin)
#if __has_builtin(__builtin_amdgcn_global_load_async_to_lds_b128)
#define USE_ASYNC_LDS 1
#else
#define USE_ASYNC_LDS 0
#endif
#if __has_builtin(__builtin_amdgcn_s_wait_asynccnt)
#define HAVE_WAIT_ASYNC_BUILTIN 1
#else
#define HAVE_WAIT_ASYNC_BUILTIN 0
#endif
#else
#define USE_ASYNC_LDS 0
#define HAVE_WAIT_ASYNC_BUILTIN 0
#endif

static constexpr int kB   = 2048;
static constexpr int kC   = 32000;
static constexpr int kNT  = 1024;      // threads per block (32 waves on wave32)
static constexpr int kNV4 = kC / 4;    // 8000 float4 per row
static constexpr int kNW  = kNT / 32;  // 32 waves

__device__ inline float wave_sum(float v) {
  #pragma unroll
  for (int off = 16; off; off >>= 1) v += __shfl_xor(v, off, 32);
  return v;
}

__global__ __launch_bounds__(kNT)
void kd_row_kernel(const float* __restrict__ out_s,
                   const float* __restrict__ out_t,
                   const int*   __restrict__ target,
                   float* __restrict__ acc /* acc[0]=kl_sum, acc[1]=count */) {
  extern __shared__ float lt[];  // kC floats: teacher row staged in LDS (128 KB)
  __shared__ float s_r0[kNW], s_r1[kNW], s_r2[kNW];
  __shared__ int   s_ix[kNW];
  __shared__ float s_mt, s_invT;
  __shared__ int   s_pred;

  const int row  = blockIdx.x;
  const int tid  = threadIdx.x;
  const int lane = tid & 31;
  const int wave = tid >> 5;

  const float4* gt4 = reinterpret_cast<const float4*>(out_t + (size_t)row * kC);
  const float4* gs4 = reinterpret_cast<const float4*>(out_s + (size_t)row * kC);
  float4* lt4 = reinterpret_cast<float4*>(lt);

  // ---- Stage teacher row into LDS via CDNA5 async DMA (ASYNCcnt path) ----
  for (int i = tid; i < kNV4; i += kNT) {
#if USE_ASYNC_LDS
    __builtin_amdgcn_global_load_async_to_lds_b128(
        (ASPC(1) v4i*)(gt4 + i), (ASPC(3) v4i*)(lt4 + i), 0, 0);
#else
    lt4[i] = gt4[i];
#endif
  }
#if USE_ASYNC_LDS
#if HAVE_WAIT_ASYNC_BUILTIN
  __builtin_amdgcn_s_wait_asynccnt(0);
#else
  asm volatile("s_wait_asynccnt 0" ::: "memory");
#endif
#endif
  __syncthreads();

  // ---- Pass A (LDS): row max of t, argmax, and sum(exp(t)) in one sweep ----
  float mx  = -3.4e38f;
  int   mi  = 0x7fffffff;
  float se0 = 0.f;
  for (int i = tid; i < kNV4; i += kNT) {
    float4 t = lt4[i];
    int base = i << 2;
    se0 += __expf(t.x) + __expf(t.y) + __expf(t.z) + __expf(t.w);
    if (t.x > mx) { mx = t.x; mi = base;     }
    if (t.y > mx) { mx = t.y; mi = base + 1; }
    if (t.z > mx) { mx = t.z; mi = base + 2; }
    if (t.w > mx) { mx = t.w; mi = base + 3; }
  }
  se0 = wave_sum(se0);
  #pragma unroll
  for (int off = 16; off; off >>= 1) {
    float ov = __shfl_xor(mx, off, 32);
    int   oi = __shfl_xor(mi, off, 32);
    if (ov > mx || (ov == mx && oi < mi)) { mx = ov; mi = oi; }
  }
  if (lane == 0) { s_r0[wave] = se0; s_r1[wave] = mx; s_ix[wave] = mi; }
  __syncthreads();
  if (wave == 0) {
    float sm = s_r0[lane];
    float v  = s_r1[lane];
    int   ix = s_ix[lane];
    sm = wave_sum(sm);
    #pragma unroll
    for (int off = 16; off; off >>= 1) {
      float ov = __shfl_xor(v, off, 32);
      int   oi = __shfl_xor(ix, off, 32);
      if (ov > v || (ov == v && oi < ix)) { v = ov; ix = oi; }
    }
    if (lane == 0) {
      s_mt   = v;
      s_pred = ix;
      float max1 = __expf(v) / sm;             // == max(softmax(out_t))
      s_invT = (max1 > 0.4f) ? 0.25f : 0.5f;   // T1=4 : T2=2
    }
  }
  __syncthreads();
  const float mt   = s_mt;
  const float invT = s_invT;

  // ---- Pass C: t from LDS, s streamed once from HBM ----
  // e = exp((t-mt)/T); A = sum e; W = sum e*(t-s); S = sum exp(s/T)
  float Aacc = 0.f, Wacc = 0.f, Sacc = 0.f;
  for (int i = tid; i < kNV4; i += kNT) {
    float4 t = lt4[i];
    float4 s = gs4[i];
    float e;
    e = __expf((t.x - mt) * invT); Aacc += e; Wacc += e * (t.x - s.x); Sacc += __expf(s.x * invT);
    e = __expf((t.y - mt) * invT); Aacc += e; Wacc += e * (t.y - s.y); Sacc += __expf(s.y * invT);
    e = __expf((t.z - mt) * invT); Aacc += e; Wacc += e * (t.z - s.z); Sacc += __expf(s.z * invT);
    e = __expf((t.w - mt) * invT); Aacc += e; Wacc += e * (t.w - s.w); Sacc += __expf(s.w * invT);
  }
  Aacc = wave_sum(Aacc);
  Wacc = wave_sum(Wacc);
  Sacc = wave_sum(Sacc);
  if (lane == 0) { s_r0[wave] = Aacc; s_r1[wave] = Wacc; s_r2[wave] = Sacc; }
  __syncthreads();
  if (wave == 0) {
    float a  = wave_sum(s_r0[lane]);
    float w  = wave_sum(s_r1[lane]);
    float sS = wave_sum(s_r2[lane]);
    if (lane == 0) {
      if (target[row] == s_pred) {
        // KL(q || softmax(s/T)) = (W/A - mt)/T + log(S/A)
        float kl = (w / a - mt) * invT + __logf(sS / a);
        atomicAdd(&acc[0], kl);
        atomicAdd(&acc[1], 1.0f);
      }
    }
  }
}

__global__ void kd_init(float* acc) {
  if (threadIdx.x < 2) acc[threadIdx.x] = 0.0f;
}

__global__ void kd_finalize(const float* __restrict__ acc, float* __restrict__ out) {
  out[0] = acc[0] * 16.0f / acc[1];  // * T1^2 / count
}

extern "C" void kernel_launch(void* const* d_in, const int* in_sizes, int n_in,
                              void* d_out, int out_size, void* d_ws, size_t ws_size,
                              hipStream_t stream) {
  (void)in_sizes; (void)n_in; (void)out_size; (void)ws_size;
  const float* out_s  = (const float*)d_in[0];
  const float* out_t  = (const float*)d_in[1];
  const int*   target = (const int*)d_in[2];   // jax default x64-off => int32
  float* acc = (float*)d_ws;

  hipLaunchKernelGGL(kd_init, dim3(1), dim3(32), 0, stream, acc);
  const size_t lds_bytes = (size_t)kC * sizeof(float);  // 128000 B teacher row
  hipLaunchKernelGGL(kd_row_kernel, dim3(kB), dim3(kNT), lds_bytes, stream,
                     out_s, out_t, target, acc);
  hipLaunchKernelGGL(kd_finalize, dim3(1), dim3(1), 0, stream, acc, (float*)d_out);
}